// Renderer_9337258902422
// MI455X (gfx1250) — compile-verified
//
#include <hip/hip_runtime.h>
#include <math.h>
#include <stdint.h>

// ---------------------------------------------------------------------------
// SG-lighting render kernel for MI455X (gfx1250, wave32).
// Memory-bound (~192 MB @ 23.3 TB/s => ~8 us floor). One thread = one pixel.
// The 672 B/pixel sg_illumination payload is staged global->LDS with gfx1250
// async-DMA (global_load_async_to_lds_b64, ASYNCcnt, non-temporal) using a
// flat lane-consecutive mapping so every 128B line is fully consumed.
// Double-buffered LDS: chunk n+1's DMA overlaps chunk n's VALU compute.
// LDS layout padded to 58 floats/pixel: 8B-aligned b64 targets and a
// bank-conflict-free (stride 58 mod 64 -> 32 distinct banks) read-back.
// ---------------------------------------------------------------------------

#define TPB     256          // 8 wave32 per block
#define SGS     24           // SGs per pixel
#define SGC     8            // SGs per LDS chunk
#define NCHUNK  3
#define PIXF    56           // floats per pixel per chunk (8 SG * 7)
#define PIXS    58           // padded floats: 8B-aligned + conflict-free stride
#define SGF     (SGS * 7)    // 168 floats per pixel total

#define PI_F     3.14159265358979323846f
#define TWO_PI_F 6.28318530717958647692f

__device__ __forceinline__ float clamp01f(float x) {
    return fminf(fmaxf(x, 0.0f), 1.0f);
}

// Issue one chunk's async global->LDS DMA (28 b64 elements per thread's share;
// lane-consecutive flat mapping => fully coalesced 8B/lane bursts).
__device__ __forceinline__ void stage_chunk(const float* __restrict__ sg,
                                            int blockStart, int B, int s0,
                                            uint32_t lds_base, int tid)
{
    const float* gchunk = sg + (size_t)blockStart * SGF + (size_t)(s0 * 7);
    #pragma unroll
    for (int i = 0; i < PIXF / 2; ++i) {
        int e  = tid + i * TPB;                 // 0 .. 7167
        int pl = e / 28;                        // local pixel 0..255
        int j2 = e - pl * 28;                   // b64 element within pixel
        const float* ga = gchunk + pl * SGF + 2 * j2;
        if (blockStart + pl >= B) ga = sg;      // clamp (B % 256 != 0 safety)
        uint32_t lo = lds_base + (uint32_t)(pl * PIXS + 2 * j2) * 4u;
        asm volatile("global_load_async_to_lds_b64 %0, %1, off th:TH_LOAD_NT"
                     :: "v"(lo), "v"(ga)
                     : "memory");
    }
}

__global__ __launch_bounds__(TPB) void sg_render_kernel(
    const float* __restrict__ sg,        // (B, 24, 7)
    const float* __restrict__ basecolor, // (B, 3)
    const float* __restrict__ metallic,  // (B, 1)
    const float* __restrict__ roughness, // (B, 1)
    const float* __restrict__ normal_in, // (B, 3)
    const float* __restrict__ view_in,   // (B, 3)
    float* __restrict__ out,             // (B, 3)
    int B)
{
    __shared__ float smem[2][TPB * PIXS];   // ~116 KB (double buffer)

    const int tid        = threadIdx.x;
    const int blockStart = blockIdx.x * TPB;
    const int p          = blockStart + tid;
    const bool valid     = (p < B);
    const int  pc        = valid ? p : (B - 1);

    const uint32_t base0 = (uint32_t)(size_t)(const void*)&smem[0][0];
    const uint32_t base1 = (uint32_t)(size_t)(const void*)&smem[1][0];

    // Kick off chunk 0 DMA before doing any per-pixel scalar math.
    stage_chunk(sg, blockStart, B, 0, base0, tid);

    // ---------------- per-pixel setup (overlaps chunk-0 DMA) ---------------
    const float m     = metallic[pc];
    const float rough = roughness[pc];

    float nrm[3], vdr[3], diffc[3], specc[3];
    #pragma unroll
    for (int c = 0; c < 3; ++c) {
        float s   = basecolor[pc * 3 + c];
        float lin = (s <= 0.04045f)
                  ? s * (1.0f / 12.92f)
                  : __powf((fmaxf(s, 0.04045f) + 0.055f) * (1.0f / 1.055f), 2.4f);
        diffc[c] = lin * (1.0f - m) * (1.0f / PI_F);        // diffuse / PI
        specc[c] = 0.04f * (1.0f - m) + lin * m;            // mix(0.04, lin, m)
        nrm[c]   = normal_in[pc * 3 + c];
        vdr[c]   = view_in[pc * 3 + c];
    }
    // normal = where(normal == 0, view_dirs, normal)  (componentwise, raw view)
    #pragma unroll
    for (int c = 0; c < 3; ++c)
        nrm[c] = (nrm[c] == 0.0f) ? vdr[c] : nrm[c];

    // normalize(normal), normalize(view)  (safe_sqrt semantics)
    {
        float d  = nrm[0]*nrm[0] + nrm[1]*nrm[1] + nrm[2]*nrm[2];
        float iv = __builtin_amdgcn_rsqf(fmaxf(d, 1e-8f));
        nrm[0] *= iv; nrm[1] *= iv; nrm[2] *= iv;
        d  = vdr[0]*vdr[0] + vdr[1]*vdr[1] + vdr[2]*vdr[2];
        iv = __builtin_amdgcn_rsqf(fmaxf(d, 1e-8f));
        vdr[0] *= iv; vdr[1] *= iv; vdr[2] *= iv;
    }

    // distribution_term + sg_wrap_distribution (all per-pixel invariant)
    const float a2  = fminf(fmaxf(rough * rough, 0.001f), 1.0f);
    const float amp = 1.0f / (PI_F * a2);                          // tmp != 0
    const float shE = fminf(fmaxf(2.0f / fmaxf(a2, 1e-6f), 0.5f), 30.0f);
    const float dv  = nrm[0]*vdr[0] + nrm[1]*vdr[1] + nrm[2]*vdr[2];

    float wd[3];  // warp_dir = normalize(reflect(-v, n))
    {
        float wa0 = 2.0f * dv * nrm[0] - vdr[0];
        float wa1 = 2.0f * dv * nrm[1] - vdr[1];
        float wa2 = 2.0f * dv * nrm[2] - vdr[2];
        float d   = wa0*wa0 + wa1*wa1 + wa2*wa2;
        float iv  = __builtin_amdgcn_rsqf(fmaxf(d, 1e-8f));
        wd[0] = wa0 * iv; wd[1] = wa1 * iv; wd[2] = wa2 * iv;
    }
    // warped sharpness, clamped by extract_sg inside sg_inner_product
    const float s1 = fminf(fmaxf(
        shE / (4.0f * fminf(fmaxf(dv, 0.0001f), 1.0f)), 0.5f), 30.0f);

    const float ndl = clamp01f(nrm[0]*wd[0] + nrm[1]*wd[1] + nrm[2]*wd[2]);
    const float ndv = clamp01f(dv);

    float ldh;
    {
        float h0 = wd[0] + vdr[0], h1 = wd[1] + vdr[1], h2 = wd[2] + vdr[2];
        float d  = h0*h0 + h1*h1 + h2*h2;
        float iv = __builtin_amdgcn_rsqf(fmaxf(d, 1e-8f));
        ldh = clamp01f((wd[0]*h0 + wd[1]*h1 + wd[2]*h2) * iv);
    }

    // G = ggx(a2, ndl) * ggx(a2, ndv)
    const float gl = 1.0f / fmaxf(ndl + sqrtf(fmaxf(a2 + (1.0f - a2)*ndl*ndl, 1e-8f)), 1e-6f);
    const float gv = 1.0f / fmaxf(ndv + sqrtf(fmaxf(a2 + (1.0f - a2)*ndv*ndv, 1e-8f)), 1e-6f);
    const float G  = gl * gv;

    // Fresnel with (1-ldh)^5, folded with 2pi*|ndf amp|*G*ndl
    float KF[3];
    {
        float t  = 1.0f - ldh;
        float t2 = t * t;
        float p5 = t2 * t2 * t;
        float Kspec = TWO_PI_F * amp * G * ndl;
        #pragma unroll
        for (int c = 0; c < 3; ++c)
            KF[c] = Kspec * (specc[c] + (1.0f - specc[c]) * p5);
    }

    float acc0 = 0.0f, acc1 = 0.0f, acc2 = 0.0f;

    // Wait for chunk 0, then enter the software-pipelined chunk loop.
    asm volatile("s_wait_asynccnt 0" ::: "memory");
    __syncthreads();

    for (int chunk = 0; chunk < NCHUNK; ++chunk) {
        // Prefetch next chunk into the other buffer (overlaps compute below).
        if (chunk + 1 < NCHUNK)
            stage_chunk(sg, blockStart, B, (chunk + 1) * SGC,
                        (chunk & 1) ? base0 : base1, tid);

        const float* q = &smem[chunk & 1][tid * PIXS];
        #pragma unroll
        for (int s = 0; s < SGC; ++s) {
            const float* g = q + s * 7;
            float A0 = fabsf(g[0]), A1 = fabsf(g[1]), A2v = fabsf(g[2]);
            float ax = g[3], ay = g[4], az = g[5];
            {
                float d  = ax*ax + ay*ay + az*az;
                float iv = __builtin_amdgcn_rsqf(fmaxf(d, 1e-8f));
                ax *= iv; ay *= iv; az *= iv;
            }
            float sh = fminf(fmaxf(g[6], 0.5f), 30.0f);
            float rl = __builtin_amdgcn_rcpf(sh);           // sh >= 0.5

            // ---- diffuse (SG diffuse approximation) ----
            float mudn  = clamp01f(ax*nrm[0] + ay*nrm[1] + az*nrm[2]);
            float eml   = __expf(-sh);
            float em2l  = eml * eml;
            float scale = 1.0f + 2.0f * em2l - rl;
            float bias  = (eml - em2l) * rl - em2l;
            float x     = sqrtf(fmaxf(1.0f - scale, 1e-8f));
            float x0    = 0.36f * mudn;
            float x1    = (1.0f / (4.0f * 0.36f)) * x;
            float n     = x0 + x1;
            float y     = (fabsf(x0) <= x1) ? n * (n / fmaxf(x, 1e-6f)) : mudn;
            float res   = scale * y + bias;
            float ibase = TWO_PI_F * rl * (1.0f - __expf(-2.0f * sh));
            float dterm = res * ibase;

            // ---- specular: sg_inner_product(warped_ndf, sg) ----
            float ux = s1 * wd[0] + sh * ax;
            float uy = s1 * wd[1] + sh * ay;
            float uz = s1 * wd[2] + sh * az;
            float ul = sqrtf(fmaxf(ux*ux + uy*uy + uz*uz, 1e-8f));
            float et = __expf(fminf(ul - s1 - sh, 87.5f));
            float ot = 1.0f - __expf(-2.0f * ul);
            float Db = et * ot * __builtin_amdgcn_rcpf(fmaxf(ul, 1e-6f));

            acc0 += dterm * A0  * diffc[0] + fmaxf(Db * A0  * KF[0], 0.0f);
            acc1 += dterm * A1  * diffc[1] + fmaxf(Db * A1  * KF[1], 0.0f);
            acc2 += dterm * A2v * diffc[2] + fmaxf(Db * A2v * KF[2], 0.0f);
        }

        // (a) ensure next-chunk DMA landed, (b) fence buffer reuse block-wide.
        asm volatile("s_wait_asynccnt 0" ::: "memory");
        __syncthreads();
    }

    if (valid) {
        __builtin_nontemporal_store(fmaxf(acc0, 0.0f), &out[p * 3 + 0]);
        __builtin_nontemporal_store(fmaxf(acc1, 0.0f), &out[p * 3 + 1]);
        __builtin_nontemporal_store(fmaxf(acc2, 0.0f), &out[p * 3 + 2]);
    }
}

extern "C" void kernel_launch(void* const* d_in, const int* in_sizes, int n_in,
                              void* d_out, int out_size, void* d_ws, size_t ws_size,
                              hipStream_t stream) {
    const float* sg_ill = (const float*)d_in[0];  // (B,24,7)
    const float* bcol   = (const float*)d_in[1];  // (B,3)
    const float* metal  = (const float*)d_in[2];  // (B,1)
    const float* rough  = (const float*)d_in[3];  // (B,1)
    const float* norm   = (const float*)d_in[4];  // (B,3)
    /* d_in[5] = alpha: unused by reference */
    const float* vdirs  = (const float*)d_in[6];  // (B,3)
    float* out = (float*)d_out;

    const int B = in_sizes[2];                    // metallic count == B
    const int blocks = (B + TPB - 1) / TPB;
    sg_render_kernel<<<blocks, TPB, 0, stream>>>(sg_ill, bcol, metal, rough,
                                                 norm, vdirs, out, B);
}